// SquarePartLayer_807453852263
// MI455X (gfx1250) — compile-verified
//
#include <hip/hip_runtime.h>

typedef __attribute__((ext_vector_type(2))) float v2f;
typedef __attribute__((ext_vector_type(4))) float v4f;
typedef __attribute__((ext_vector_type(8))) float v8f;

#define NB      64                 // N_BASIS (K of the GEMM)
#define BATCH   8                  // B (M of the GEMM, padded to 16)
#define NCH     4                  // C channels
#define DIM     (1u << 20)         // 4*64*64*64 columns (N of the GEMM)
#define GRES    128
#define VOXELS  ((size_t)NCH * GRES * GRES * GRES)   // per-batch grid = 8,388,608

// ---------------------------------------------------------------------------
// Kernel 1: zero the whole [8,4,128,128,128] grid with 128-bit stores.
// 67,108,864 floats = 16,777,216 float4 stores.
// ---------------------------------------------------------------------------
__global__ __launch_bounds__(256) void zero_grid(v4f* __restrict__ out) {
    size_t i = (size_t)blockIdx.x * blockDim.x + threadIdx.x;
    v4f zv = {0.0f, 0.0f, 0.0f, 0.0f};
    out[i] = zv;
}

// ---------------------------------------------------------------------------
// Kernel 2: (L*z) @ U + mu with V_WMMA_F32_16X16X4_F32, then scatter into the
// grid block. One wave handles 64 consecutive columns of U (4 N-tiles of 16),
// chaining 16 WMMAs (K=64) per accumulator. U is read exactly once.
// ---------------------------------------------------------------------------
__global__ __launch_bounds__(256) void subspace_wmma(
    const float* __restrict__ z,   // [8, 64]
    const float* __restrict__ U,   // [64, DIM] row-major
    const float* __restrict__ L,   // [64]
    const float* __restrict__ mu,  // [DIM]
    float* __restrict__ out)       // [8, 4, 128, 128, 128]
{
    const int lane = threadIdx.x & 31;
    const int m    = lane & 15;          // M index (A) / N index (B) per layout
    const int hi   = lane >> 4;          // half-wave selector (K offset 0 or 2)
    const size_t wave    = (size_t)blockIdx.x * 8 + (threadIdx.x >> 5);
    const size_t colBase = wave * 64;

    // --- Build A operand: Lz is 16x64 f32, rows 8..15 are zero padding. ---
    // 16x4 f32 A layout: lanes 0-15 hold M=0-15; VGPR0 = K={0|2}+base,
    // VGPR1 = K={1|3}+base, selected by half-wave.
    const int  mm    = (m < BATCH) ? m : 0;
    const bool valid = (m < BATCH);
    v2f a[16];
#pragma unroll
    for (int t = 0; t < 16; ++t) {
        const int k0 = 4 * t + 2 * hi;
        const float lz0 = L[k0]     * z[mm * NB + k0];
        const float lz1 = L[k0 + 1] * z[mm * NB + k0 + 1];
        a[t].x = valid ? lz0 : 0.0f;
        a[t].y = valid ? lz1 : 0.0f;
    }

#pragma unroll
    for (int nt = 0; nt < 4; ++nt) {
        const size_t n0 = colBase + (size_t)nt * 16;
        v8f c = {};
        // --- K reduction: 16 x V_WMMA_F32_16X16X4_F32 ---
        // 4x16 f32 B layout mirrors A: lanes 0-15 = N, VGPR pair = rows K,K+1
        // (half-wave offset +2). Each half-wave load is 16 consecutive floats.
#pragma unroll
        for (int t = 0; t < 16; ++t) {
            const int    krow = 4 * t + 2 * hi;
            const size_t col  = n0 + (size_t)m;
            v2f b;
            b.x = U[(size_t)krow       * DIM + col];
            b.y = U[(size_t)(krow + 1) * DIM + col];
            c = __builtin_amdgcn_wmma_f32_16x16x4_f32(
                    /*neg_a=*/false, a[t], /*neg_b=*/false, b,
                    /*c_mod=*/(short)0, c, /*reuse_a=*/false, /*reuse_b=*/false);
        }

        // --- Scatter: lanes 0-15 hold batches 0..7 in c[0..7] for column n. ---
        if (lane < 16) {
            const size_t n   = n0 + (size_t)lane;
            const float  muv = mu[n];
            const int kk = (int)(n & 63);          // -> w - 32
            const int jj = (int)((n >> 6) & 63);   // -> h - 32
            const int ii = (int)((n >> 12) & 63);  // -> d (split)
            const int cc = (int)((n >> 18) & 3);   // channel
            const int d  = (ii < 32) ? ii : ii + 64;
            const size_t base =
                (((size_t)cc * GRES + d) * GRES + (jj + 32)) * GRES + (kk + 32);
#pragma unroll
            for (int bb = 0; bb < BATCH; ++bb)
                out[(size_t)bb * VOXELS + base] = c[bb] + muv;
        }
    }
}

// ---------------------------------------------------------------------------
extern "C" void kernel_launch(void* const* d_in, const int* in_sizes, int n_in,
                              void* d_out, int out_size, void* d_ws, size_t ws_size,
                              hipStream_t stream) {
    const float* z  = (const float*)d_in[0];
    const float* U  = (const float*)d_in[1];
    const float* L  = (const float*)d_in[2];
    const float* mu = (const float*)d_in[3];
    float* out = (float*)d_out;

    // 67,108,864 floats -> 16,777,216 float4 -> 65536 blocks x 256 threads.
    zero_grid<<<65536, 256, 0, stream>>>((v4f*)out);

    // DIM/64 = 16384 waves; 8 waves per 256-thread block -> 2048 blocks.
    subspace_wmma<<<2048, 256, 0, stream>>>(z, U, L, mu, out);
}